// Scratches_58385785422324
// MI455X (gfx1250) — compile-verified
//
#include <hip/hip_runtime.h>
#include <cstdint>

// ---------------------------------------------------------------------------
// out = img (32x3x512x512 f32) then scatter 1.0 along 640 short lines
// (32 images x 20 scratches x 75 steps, thickened +1 in x and +1 in y).
// Memory-bound: ~192 MB traffic -> ~8.3 us floor at 23.3 TB/s. No matrix ops,
// so the CDNA5-specific path is the async global<->LDS DMA engine
// (ASYNCcnt-tracked b128 transfers, GVS addressing: saddr base + one shared
// 32-bit voffset + offset: immediates; the instruction offset is added to
// both the global and LDS addresses, so one LDS base serves all 8 slots).
// ---------------------------------------------------------------------------

#define COPY_BLOCK 256
#define CHUNKS_PER_THREAD 8   // 8 x float4 = 128 B per thread
// per-block bytes = 256 * 8 * 16 = 32768; per-k stride = 256 * 16 = 4096 B

// Generic pointer to a __shared__ object is (SHARED_BASE<<32 | lds_offset);
// the async instructions take the raw 32-bit LDS byte address.
__device__ __forceinline__ uint32_t lds_addr32(const void* p) {
  return (uint32_t)(uintptr_t)p;
}

__device__ __forceinline__ void wait_async0() {
#if defined(__AMDGCN__) && __has_builtin(__builtin_amdgcn_s_wait_asynccnt)
  __builtin_amdgcn_s_wait_asynccnt(0);
#else
  asm volatile("s_wait_asynccnt 0" ::: "memory");
#endif
}

// Bulk copy: each block moves exactly COPY_BLOCK*CHUNKS_PER_THREAD float4
// chunks through a 32 KB LDS staging buffer via async DMA. Grid is sized so
// no bounds checks are needed (remainder handled by a separate tail kernel).
__global__ void __launch_bounds__(COPY_BLOCK)
scratches_copy_async(const float4* __restrict__ in, float4* __restrict__ out) {
  __shared__ float4 buf[COPY_BLOCK * CHUNKS_PER_THREAD];  // 32 KB
  const uint32_t t = threadIdx.x;
  // one shared 32-bit byte offset (max ~96 MB, fits signed i32)
  const uint32_t goff =
      blockIdx.x * (uint32_t)(COPY_BLOCK * CHUNKS_PER_THREAD * 16) + t * 16u;
  const uint32_t vlds = lds_addr32(&buf[t]);  // slot k lives at vlds + k*4096
  const unsigned long long gin  = (unsigned long long)in;
  const unsigned long long gout = (unsigned long long)out;

#define ALD(OFF)                                                           \
  asm volatile("global_load_async_to_lds_b128 %0, %1, %2 offset:" #OFF     \
               :: "v"(vlds), "v"(goff), "s"(gin) : "memory")
#define AST(OFF)                                                           \
  asm volatile("global_store_async_from_lds_b128 %0, %1, %2 offset:" #OFF  \
               :: "v"(goff), "v"(vlds), "s"(gout) : "memory")

  ALD(0); ALD(4096); ALD(8192); ALD(12288);
  ALD(16384); ALD(20480); ALD(24576); ALD(28672);
  wait_async0();  // loads landed in LDS; prior to this, 8 loads in flight
  AST(0); AST(4096); AST(8192); AST(12288);
  AST(16384); AST(20480); AST(24576); AST(28672);
#undef ALD
#undef AST
  // outstanding stores drained by the implicit wait-idle at s_endpgm
}

// Tail: plain vector copy for any chunks not covered by full blocks
// (zero for the reference shape; kept for robustness).
__global__ void scratches_copy_tail(const float4* __restrict__ in,
                                    float4* __restrict__ out,
                                    long long start, long long nchunks) {
  long long i = start + (long long)blockIdx.x * blockDim.x + threadIdx.x;
  if (i < nchunks) out[i] = in[i];
}

// One thread per (image n, scratch s, step i). Matches reference math:
// endpoints = start + length*(cos,sin)(angle); lerp; truncate to int; clamp;
// write 1.0 to (y,x), (y+1,x), (y,x+1) across all channels.
__global__ void scratches_draw(float* __restrict__ out,
                               const float* __restrict__ angles,
                               const float* __restrict__ lengths,
                               const int* __restrict__ x_start,
                               const int* __restrict__ y_start,
                               int N, int S, int T, int C, int H, int W) {
  int id = blockIdx.x * blockDim.x + threadIdx.x;
  int total = N * S * T;
  if (id >= total) return;

  int i = id % T;
  int ns = id / T;
  int s = ns % S;
  int n = ns / S;

  int ss = n * S + s;
  float ang = angles[ss];
  float len = lengths[ss];
  float xs = (float)x_start[ss];
  float ys = (float)y_start[ss];
  float xe = xs + len * cosf(ang);
  float ye = ys + len * sinf(ang);

  float tt = (float)i / (float)(T - 1);   // linspace(0,1,T)
  float fx = xs * (1.0f - tt) + xe * tt;
  float fy = ys * (1.0f - tt) + ye * tt;

  int x = (int)fx;                         // truncate toward zero (== .astype(int32))
  int y = (int)fy;
  x = min(max(x, 0), W - 1);
  y = min(max(y, 0), H - 1);
  int x1 = min(x + 1, W - 1);
  int y1 = min(y + 1, H - 1);

  long long plane = (long long)H * W;
  long long nb = (long long)n * C * plane;
#pragma unroll
  for (int c = 0; c < 3; ++c) {
    long long cb = nb + (long long)c * plane;
    out[cb + (long long)y  * W + x ] = 1.0f;
    out[cb + (long long)y1 * W + x ] = 1.0f;
    out[cb + (long long)y  * W + x1] = 1.0f;
  }
}

extern "C" void kernel_launch(void* const* d_in, const int* in_sizes, int n_in,
                              void* d_out, int out_size, void* d_ws, size_t ws_size,
                              hipStream_t stream) {
  const float* img     = (const float*)d_in[0];
  const float* angles  = (const float*)d_in[1];
  const float* lengths = (const float*)d_in[2];
  const int*   xst     = (const int*)d_in[3];
  const int*   yst     = (const int*)d_in[4];
  float* out = (float*)d_out;

  const int S = 20, T = 75, C = 3, H = 512, W = 512;
  const int N = in_sizes[1] / S;                       // 32

  // ---- bulk copy (async DMA through LDS), guard-free full blocks ----
  long long nchunks   = (long long)in_sizes[0] / 4;    // 6,291,456 float4 chunks
  long long per_block = (long long)COPY_BLOCK * CHUNKS_PER_THREAD;  // 2048
  long long full      = nchunks / per_block;           // 3072 (exact here)
  if (full > 0) {
    scratches_copy_async<<<(int)full, COPY_BLOCK, 0, stream>>>(
        (const float4*)img, (float4*)out);
  }
  long long rem = nchunks - full * per_block;          // 0 for reference shape
  if (rem > 0) {
    scratches_copy_tail<<<(int)((rem + 255) / 256), 256, 0, stream>>>(
        (const float4*)img, (float4*)out, full * per_block, nchunks);
  }

  // ---- scatter the scratches ----
  int total = N * S * T;                               // 48,000
  scratches_draw<<<(total + 255) / 256, 256, 0, stream>>>(
      out, angles, lengths, xst, yst, N, S, T, C, H, W);
}